// GraphLayer_dgcnn_3513283248939
// MI455X (gfx1250) — compile-verified
//
#include <hip/hip_runtime.h>
#include <hip/hip_bf16.h>

// DGCNN graph feature layer for MI455X (gfx1250, wave32).
// B=16, C=64, N=2048, K=20, K2=14. Output = feature(16,128,2048,20) ++ idx_flat(16*2048*20).
//
// Roofline: Gram matmul is only ~8.6 GFLOP; output stores are 335MB -> the op is
// bandwidth/top-k bound. Therefore fp32 WMMA (16x16x4) is used: exact reference
// semantics at zero throughput cost, while using the CDNA5 matrix pipe.

#define BB 16
#define CC 64
#define NN 2048
#define KK 20
#define NREM 6            // 20 - ceil(20*2/3) = 6 smallest removed
#define ROWS 16           // rows per knn workgroup
#define DSTRIDE 2052      // 2048 + 4 pad: kills LDS bank conflicts on tile stores
#define FEAT_TOTAL (BB * 2 * CC * NN * KK)   // 83,886,080

typedef __attribute__((ext_vector_type(2))) float v2f;
typedef __attribute__((ext_vector_type(8))) float v8f;

// ---------------------------------------------------------------------------
// Kernel 1: xt[b][n][c] = x[b][c][n];  xx[b][n] = sum_c x^2
// ---------------------------------------------------------------------------
__global__ void prep_kernel(const float* __restrict__ x,
                            float* __restrict__ xt,
                            float* __restrict__ xx) {
  int i = blockIdx.x * blockDim.x + threadIdx.x;   // point id in [0, B*N)
  if (i >= BB * NN) return;
  int b = i / NN, n = i % NN;
  const float* xp = x + (size_t)b * CC * NN + n;   // stride-N reads, coalesced across lanes
  float* xo = xt + (size_t)i * CC;
  float s = 0.f;
#pragma unroll
  for (int c = 0; c < CC; ++c) {
    float v = xp[(size_t)c * NN];
    s += v * v;
    xo[c] = v;
  }
  xx[i] = s;
}

// ---------------------------------------------------------------------------
// Kernel 2: per (b, 16-row block): fp32-WMMA distance tile into LDS, then
// wave-cooperative top-20 per row.  Dynamic LDS: 16 * DSTRIDE * 4 = 131,328 B.
// ---------------------------------------------------------------------------
__global__ void knn_kernel(const float* __restrict__ xt,
                           const float* __restrict__ xx,
                           int* __restrict__ idx_out,
                           float* __restrict__ out) {
  extern __shared__ float dist[];                  // [ROWS][DSTRIDE]

  const int wave = threadIdx.x >> 5;
  const int lane = threadIdx.x & 31;
  const int half = lane >> 4;                      // 0: lanes 0-15, 1: lanes 16-31
  const int mn   = lane & 15;

  const int b        = blockIdx.x >> 7;            // 128 row-blocks per batch
  const int row_base = (blockIdx.x & 127) << 4;

  const float* xtb = xt + (size_t)b * NN * CC;
  const float* xxb = xx + (size_t)b * NN;

  // ---- A fragments: rows row_base..row_base+15, all 64 K values ----------
  // fp32 16x4 A layout (ISA 7.12.2): lanes 0-15 hold K={0,1}, lanes 16-31 K={2,3}.
  v2f afrag[16];
#pragma unroll
  for (int ks = 0; ks < 16; ++ks) {
    const float* p = xtb + (size_t)(row_base + mn) * CC + (ks * 4 + half * 2);
    afrag[ks] = *reinterpret_cast<const v2f*>(p);
  }
  float xxr[8];
#pragma unroll
  for (int r = 0; r < 8; ++r) xxr[r] = xxb[row_base + half * 8 + r];

  // ---- distance tiles: 128 column tiles split over 8 waves ----------------
  for (int t = wave; t < NN / 16; t += 8) {
    const int col_base = t * 16;
    v2f bfrag[16];
#pragma unroll
    for (int ks = 0; ks < 16; ++ks) {
      const float* p = xtb + (size_t)(col_base + mn) * CC + (ks * 4 + half * 2);
      bfrag[ks] = *reinterpret_cast<const v2f*>(p);
    }
    v8f acc = {};
#pragma unroll
    for (int ks = 0; ks < 16; ++ks) {
      acc = __builtin_amdgcn_wmma_f32_16x16x4_f32(
          /*neg_a=*/false, afrag[ks], /*neg_b=*/false, bfrag[ks],
          /*c_mod=*/(short)0, acc, /*reuse_a=*/false, /*reuse_b=*/false);
    }
    const float xxc = xxb[col_base + mn];
    // C/D layout: VGPR r -> (M = half*8 + r, N = mn)
#pragma unroll
    for (int r = 0; r < 8; ++r) {
      float pd = 2.0f * acc[r] - xxr[r] - xxc;     // -(|xi|^2 - 2 xi.xj + |xj|^2)
      dist[(half * 8 + r) * DSTRIDE + col_base + mn] = pd;
    }
  }
  __syncthreads();

  // ---- top-20 per row: wave handles rows 2*wave and 2*wave+1 --------------
  for (int rr = 0; rr < 2; ++rr) {
    const int rloc = wave * 2 + rr;
    const int row  = row_base + rloc;

    // lane-local sorted (desc) top-20 over 64 strided candidates
    float vals[KK];
    int   cols[KK];
#pragma unroll
    for (int i = 0; i < KK; ++i) { vals[i] = -INFINITY; cols[i] = 0x7FFFFFFF; }

    for (int t = 0; t < NN / 32; ++t) {
      const int c = lane + 32 * t;
      float v = dist[rloc * DSTRIDE + c];
      int   ci = c;
      if (v > vals[KK - 1] || (v == vals[KK - 1] && ci < cols[KK - 1])) {
#pragma unroll
        for (int i = 0; i < KK; ++i) {
          bool pred = (v > vals[i]) || (v == vals[i] && ci < cols[i]);
          float nv = pred ? vals[i] : v;
          int   nc = pred ? cols[i] : ci;
          vals[i] = pred ? v : vals[i];
          cols[i] = pred ? ci : cols[i];
          v = nv; ci = nc;
        }
      }
    }

    // 20-step wave32 merge: shuffle-argmax of list heads (val desc, col asc)
    for (int q = 0; q < KK; ++q) {
      float hv = vals[0];
      int   hc = cols[0];
#pragma unroll
      for (int off = 16; off > 0; off >>= 1) {
        float ov = __shfl_xor(hv, off, 32);
        int   oc = __shfl_xor(hc, off, 32);
        if (ov > hv || (ov == hv && oc < hc)) { hv = ov; hc = oc; }
      }
      if (lane == 0) {
        size_t o = ((size_t)b * NN + row) * KK + q;
        idx_out[o] = hc;                                  // local n within batch
        out[(size_t)FEAT_TOTAL + o] = (float)(hc + b * NN);  // idx_flat tail
      }
      if (lane == (hc & 31)) {                            // owner pops its head
#pragma unroll
        for (int i = 0; i < KK - 1; ++i) { vals[i] = vals[i + 1]; cols[i] = cols[i + 1]; }
        vals[KK - 1] = -INFINITY;
        cols[KK - 1] = 0x7FFFFFFF;
      }
    }
  }
}

// ---------------------------------------------------------------------------
// Kernel 3: x1[p][c] = (sum of 20 gathered neighbor values - 6 smallest) / 14
// Block = 256 threads = 4 points x 64 channels.
// ---------------------------------------------------------------------------
__global__ void smooth_kernel(const float* __restrict__ xt,
                              const int* __restrict__ idx,
                              float* __restrict__ x1) {
  __shared__ int sidx[4 * KK];
  const int tid = threadIdx.x;
  if (tid < 4 * KK) sidx[tid] = idx[(size_t)blockIdx.x * (4 * KK) + tid];
  __syncthreads();

  const int p = blockIdx.x * 4 + (tid >> 6);       // global point id
  const int c = tid & 63;
  const int b = p / NN;
  const float* xtb = xt + (size_t)b * NN * CC;
  const int lp = tid >> 6;

  float g[KK];
  float sum = 0.f;
#pragma unroll
  for (int j = 0; j < KK; ++j) {
    g[j] = xtb[(size_t)sidx[lp * KK + j] * CC + c];
    sum += g[j];
  }
#pragma unroll
  for (int rm = 0; rm < NREM; ++rm) {
    float mv = g[0];
    int   mp = 0;
#pragma unroll
    for (int j = 1; j < KK; ++j) {
      if (g[j] < mv) { mv = g[j]; mp = j; }
    }
    sum -= mv;
#pragma unroll
    for (int j = 0; j < KK; ++j) {
      if (j == mp) g[j] = INFINITY;
    }
  }
  x1[(size_t)p * CC + c] = sum * (1.0f / 14.0f);
}

// ---------------------------------------------------------------------------
// Kernel 4: feature(b, ch, n, j):
//   ch <  64 : x1[b][idx[b,n,j]][ch] - xt[b][n][ch]
//   ch >= 64 : xt[b][n][ch-64]
// 1 thread / element; consecutive lanes vary j,n -> coalesced stores.
// ---------------------------------------------------------------------------
__global__ void feature_kernel(const float* __restrict__ xt,
                               const float* __restrict__ x1,
                               const int* __restrict__ idx,
                               float* __restrict__ out) {
  unsigned int i = blockIdx.x * 256u + threadIdx.x;   // < 83,886,080 fits u32
  const unsigned int j  = i % KK;
  const unsigned int n  = (i / KK) % NN;
  const unsigned int ch = (i / (KK * NN)) % (2 * CC);
  const unsigned int b  = i / (KK * NN * 2 * CC);

  const size_t pn = (size_t)b * NN + n;
  float v;
  if (ch < CC) {
    const int nb = idx[pn * KK + j];
    v = x1[((size_t)b * NN + nb) * CC + ch] - xt[pn * CC + ch];
  } else {
    v = xt[pn * CC + (ch - CC)];
  }
  out[i] = v;
}

// ---------------------------------------------------------------------------
extern "C" void kernel_launch(void* const* d_in, const int* in_sizes, int n_in,
                              void* d_out, int out_size, void* d_ws, size_t ws_size,
                              hipStream_t stream) {
  const float* x = (const float*)d_in[0];   // (B, C, N) fp32
  // d_in[1] = k (==20), d_in[2] = local_idx : unused by this variant
  float* out = (float*)d_out;

  // workspace layout (~18.6 MB)
  float* xt = (float*)d_ws;                        // B*N*C
  float* x1 = xt + (size_t)BB * NN * CC;           // B*N*C
  float* xx = x1 + (size_t)BB * NN * CC;           // B*N
  int*  idx = (int*)(xx + (size_t)BB * NN);        // B*N*K

  prep_kernel<<<(BB * NN + 255) / 256, 256, 0, stream>>>(x, xt, xx);

  const size_t knn_lds = (size_t)ROWS * DSTRIDE * sizeof(float);  // 131,328 B
  knn_kernel<<<BB * (NN / ROWS), 256, knn_lds, stream>>>(xt, xx, idx, out);

  smooth_kernel<<<BB * NN / 4, 256, 0, stream>>>(xt, idx, x1);

  feature_kernel<<<FEAT_TOTAL / 256, 256, 0, stream>>>(xt, x1, idx, out);
}